// Community_Convolution_layer_1949915152709
// MI455X (gfx1250) — compile-verified
//
#include <hip/hip_runtime.h>
#include <hip/hip_bf16.h>

// ---------------------------------------------------------------------------
// Community_Convolution_layer for MI455X (gfx1250, wave32, WMMA)
// Outputs (concat, fp32): W_out (10*3072*3072) then Hp_k (10*16*192*70)
// ---------------------------------------------------------------------------

namespace {

constexpr int Bn = 10;     // batch
constexpr int Kc = 16;     // communities
constexpr int NP = 192;    // nodes per community
constexpr int Nn = 3072;   // N
constexpr int FD = 70;     // feature dim
constexpr float ALPHA = 0.1f;
constexpr float BETA  = 0.1f;

constexpr int ROWS_PER_BLK = 8;                 // W_out rows per block
constexpr size_t WOUT_ELEMS = (size_t)Bn * Nn * Nn;

constexpr int FDP = 80;                         // FD padded to 16-tile multiple
constexpr int KP  = 72;                         // theta K dim padded to mult of 4

typedef __attribute__((ext_vector_type(2))) float v2f;
typedef __attribute__((ext_vector_type(4))) float v4f;
typedef __attribute__((ext_vector_type(8))) float v8f;

__device__ __forceinline__ v8f wmma4(v2f a, v2f b, v8f c) {
  // D = A(16x4,f32) x B(4x16,f32) + C(16x16,f32)
  return __builtin_amdgcn_wmma_f32_16x16x4_f32(
      /*neg_a=*/false, a, /*neg_b=*/false, b,
      /*c_mod=*/(short)0, c, /*reuse_a=*/false, /*reuse_b=*/false);
}

// ---------------------------------------------------------------------------
// Kernel 1: ratio[b,i,j] = Rct/Rc = 1 + dd[j] + Rc[j,i]*dd[i]/Rc[i,j]
//           dd[j] = ALPHA * (colsum(Hc) . Rc[:,j]) / rcs_diag[b,j,j]
// One block per batch, 256 threads = 16x16.
// ---------------------------------------------------------------------------
__global__ void ratio_kernel(const float* __restrict__ Hc,
                             const float* __restrict__ Rc,
                             const float* __restrict__ rcs,
                             float* __restrict__ ratio) {
  const int b = blockIdx.x;
  const int tid = threadIdx.x;           // 0..255
  const int r = tid >> 4, c = tid & 15;

  __shared__ float sHc[Kc * Kc];
  __shared__ float sRc[Kc * Kc];
  __shared__ float colsum[Kc];
  __shared__ float dd[Kc];

  sHc[tid] = Hc[b * Kc * Kc + tid];
  sRc[tid] = Rc[b * Kc * Kc + tid];
  __syncthreads();

  if (tid < Kc) {
    float s = 0.0f;
    for (int i = 0; i < Kc; ++i) s += sHc[i * Kc + tid];
    colsum[tid] = s;
  }
  __syncthreads();

  if (tid < Kc) {
    float s = 0.0f;
    for (int j = 0; j < Kc; ++j) s += colsum[j] * sRc[j * Kc + tid];
    const float diag = rcs[b * Kc * Kc + tid * Kc + tid];
    dd[tid] = ALPHA * s / diag;
  }
  __syncthreads();

  const float rc_ij = sRc[r * Kc + c];
  const float rc_ji = sRc[c * Kc + r];
  ratio[b * Kc * Kc + tid] = 1.0f + dd[c] + rc_ji * dd[r] / rc_ij;
}

// ---------------------------------------------------------------------------
// Kernel 2 (bandwidth-dominant): W_out = W * fac, streaming b128 NT.
// fac[b,i,j] = (memb[b,i]==memb[b,j]) ? 1 : ratio[b, memb[b,i], memb[b,j]]
// Block = (b, 8-row strip); ratio (1KB) + memb row (12KB) staged in LDS.
// ---------------------------------------------------------------------------
__global__ void wfac_kernel(const float* __restrict__ W,
                            const int* __restrict__ memb,
                            const float* __restrict__ ratio,
                            float* __restrict__ Wout) {
  const int blk  = blockIdx.x;
  const int b    = blk / (Nn / ROWS_PER_BLK);
  const int i0   = (blk % (Nn / ROWS_PER_BLK)) * ROWS_PER_BLK;
  const int tid  = threadIdx.x;          // 256 threads

  __shared__ float sRatio[Kc * Kc];
  __shared__ int   sMemb[Nn];

  sRatio[tid] = ratio[b * Kc * Kc + tid];
  const int* mb = memb + (size_t)b * Nn;
  for (int j = tid; j < Nn; j += 256) sMemb[j] = mb[j];
  __syncthreads();

  const float* Wb = W    + (size_t)b * Nn * Nn;
  float*       Wo = Wout + (size_t)b * Nn * Nn;

  for (int i = i0; i < i0 + ROWS_PER_BLK; ++i) {
    const int mi = sMemb[i];
    const float* lut = &sRatio[mi * Kc];
    const v4f* wrow = (const v4f*)(Wb + (size_t)i * Nn);
    v4f*       orow = (v4f*)(Wo + (size_t)i * Nn);
    for (int j4 = tid; j4 < Nn / 4; j4 += 256) {
      const v4f w = __builtin_nontemporal_load(&wrow[j4]);
      const int j = j4 * 4;
      const int m0 = sMemb[j + 0], m1 = sMemb[j + 1];
      const int m2 = sMemb[j + 2], m3 = sMemb[j + 3];
      v4f o;
      o.x = w.x * ((m0 == mi) ? 1.0f : lut[m0]);
      o.y = w.y * ((m1 == mi) ? 1.0f : lut[m1]);
      o.z = w.z * ((m2 == mi) ? 1.0f : lut[m2]);
      o.w = w.w * ((m3 == mi) ? 1.0f : lut[m3]);
      __builtin_nontemporal_store(o, &orow[j4]);
    }
  }
}

// ---------------------------------------------------------------------------
// Kernel 3: T1[b,k] = A @ Hp  (192x192 @ 192x70), A fused on the fly:
//   A[n,m] = Wp[n,m]*Rn[n,m]*dis[n]*dis[m],  dis = rsqrt(deg) (staged in LDS)
// Hp staged in LDS zero-padded to 80 cols -> no guards in the K loop.
// One block per (b,k); 8 waves; fp32 WMMA 16x16x4, 48 K-steps per tile.
// ---------------------------------------------------------------------------
__global__ void ahp_kernel(const float* __restrict__ Wp,
                           const float* __restrict__ Rn,
                           const float* __restrict__ Hp,
                           const float* __restrict__ deg,
                           float* __restrict__ T1) {
  const int bk = blockIdx.x;                       // 0..159
  const float* wp = Wp + (size_t)bk * NP * NP;
  const float* rn = Rn + (size_t)bk * NP * NP;
  const float* hp = Hp + (size_t)bk * NP * FD;
  const float* dg = deg + (size_t)bk * NP;
  float*       t1 = T1 + (size_t)bk * NP * FD;

  __shared__ float sdis[NP];                       // 768 B
  __shared__ float sHp[NP][FDP];                   // 60 KB, zero-padded cols

  const int tid = threadIdx.x;
  if (tid < NP) sdis[tid] = rsqrtf(dg[tid]);
  for (int idx = tid; idx < NP * FDP; idx += 256) {
    const int rr = idx / FDP, cc = idx % FDP;
    sHp[rr][cc] = (cc < FD) ? hp[(size_t)rr * FD + cc] : 0.0f;
  }
  __syncthreads();

  const int wave = tid >> 5;                       // 8 waves
  const int lane = tid & 31;
  const int half = lane >> 4;
  const int r    = lane & 15;

  constexpr int MT = NP / 16;                      // 12
  constexpr int NT = FDP / 16;                     // 5

  for (int t = wave; t < MT * NT; t += 8) {
    const int mt = t / NT, nt = t % NT;
    const int row = mt * 16 + r;                   // A row owned by lane
    const int n   = nt * 16 + r;                   // B/D column owned by lane
    const float disn = sdis[row];
    v8f c = {};
    for (int kk = 0; kk < NP / 4; ++kk) {
      const int k0 = kk * 4 + half * 2;
      // A fragment: a = {A[row,k0], A[row,k0+1]} (fused scaling)
      const v2f wv = *(const v2f*)(wp + (size_t)row * NP + k0);
      const v2f rv = *(const v2f*)(rn + (size_t)row * NP + k0);
      v2f a;
      a.x = wv.x * rv.x * disn * sdis[k0];
      a.y = wv.y * rv.y * disn * sdis[k0 + 1];
      // B fragment from zero-padded LDS: unconditional
      v2f bf;
      bf.x = sHp[k0][n];
      bf.y = sHp[k0 + 1][n];
      c = wmma4(a, bf, c);
    }
    if (n < FD) {
      for (int v = 0; v < 8; ++v) {
        const int m = mt * 16 + v + half * 8;      // D row for c[v]
        t1[(size_t)m * FD + n] = c[v];
      }
    }
  }
}

// ---------------------------------------------------------------------------
// Kernel 4: Hp_k[b,k] = BETA * T1 @ theta  (192x70 @ 70x70), fp32 WMMA.
// theta staged in LDS zero-padded to 72x80 -> unconditional B fragments.
// A fragment: clamped-index b64 load + mask multiply (no exec divergence).
// ---------------------------------------------------------------------------
__global__ void theta_kernel(const float* __restrict__ T1,
                             const float* __restrict__ theta,
                             float* __restrict__ out) {
  const int bk = blockIdx.x;                       // 0..159
  const float* t1 = T1  + (size_t)bk * NP * FD;
  float*       o  = out + (size_t)bk * NP * FD;

  __shared__ float sTh[KP][FDP];                   // 72x80 = 23 KB, zero-padded

  const int tid = threadIdx.x;
  for (int idx = tid; idx < KP * FDP; idx += 256) {
    const int rr = idx / FDP, cc = idx % FDP;
    sTh[rr][cc] = (rr < FD && cc < FD) ? theta[(size_t)rr * FD + cc] : 0.0f;
  }
  __syncthreads();

  const int wave = tid >> 5;
  const int lane = tid & 31;
  const int half = lane >> 4;
  const int r    = lane & 15;

  constexpr int MT = NP / 16;                      // 12
  constexpr int NT = FDP / 16;                     // 5
  constexpr int KS = KP / 4;                       // 18

  for (int t = wave; t < MT * NT; t += 8) {
    const int mt = t / NT, nt = t % NT;
    const int row = mt * 16 + r;
    const int n   = nt * 16 + r;
    v8f c = {};
    for (int kk = 0; kk < KS; ++kk) {
      const int k0 = kk * 4 + half * 2;            // even, max 70
      // Clamped in-bounds b64 load + {0,1} mask: no exec-mask branching.
      const int   k0c   = (k0 <= FD - 2) ? k0 : (FD - 2);
      const float kmask = (k0 <= FD - 2) ? 1.0f : 0.0f;
      const v2f tv = *(const v2f*)(t1 + (size_t)row * FD + k0c);
      v2f a;
      a.x = tv.x * kmask;
      a.y = tv.y * kmask;
      v2f bf;
      bf.x = sTh[k0][n];
      bf.y = sTh[k0 + 1][n];
      c = wmma4(a, bf, c);
    }
    if (n < FD) {
      for (int v = 0; v < 8; ++v) {
        const int m = mt * 16 + v + half * 8;
        o[(size_t)m * FD + n] = BETA * c[v];
      }
    }
  }
}

}  // namespace

// ---------------------------------------------------------------------------
// Launch: inputs in setup_inputs() order:
// 0:Hc 1:Rc 2:rcs_diag 3:Wp 4:Rn 5:Hp 6:deg 7:W 8:theta 9:memb
// ---------------------------------------------------------------------------
extern "C" void kernel_launch(void* const* d_in, const int* in_sizes, int n_in,
                              void* d_out, int out_size, void* d_ws, size_t ws_size,
                              hipStream_t stream) {
  const float* Hc    = (const float*)d_in[0];
  const float* Rc    = (const float*)d_in[1];
  const float* rcs   = (const float*)d_in[2];
  const float* Wp    = (const float*)d_in[3];
  const float* Rn    = (const float*)d_in[4];
  const float* Hp    = (const float*)d_in[5];
  const float* deg   = (const float*)d_in[6];
  const float* W     = (const float*)d_in[7];
  const float* theta = (const float*)d_in[8];
  const int*   memb  = (const int*)d_in[9];

  float* out_W  = (float*)d_out;                   // B*N*N
  float* out_Hp = out_W + WOUT_ELEMS;              // B*K*NP*FD

  // Workspace: ratio (B*256 f32, 10 KB) then T1 (B*K*NP*FD f32, 8.4 MB)
  float* ws_ratio = (float*)d_ws;
  float* ws_T1    = (float*)((char*)d_ws + 16384); // keep 16B alignment

  ratio_kernel<<<Bn, 256, 0, stream>>>(Hc, Rc, rcs, ws_ratio);
  wfac_kernel<<<Bn * (Nn / ROWS_PER_BLK), 256, 0, stream>>>(W, memb, ws_ratio, out_W);
  ahp_kernel<<<Bn * Kc, 256, 0, stream>>>(Wp, Rn, Hp, deg, ws_T1);
  theta_kernel<<<Bn * Kc, 256, 0, stream>>>(ws_T1, theta, out_Hp);
}